// FittingDynamic_65300682768679
// MI455X (gfx1250) — compile-verified
//
#include <hip/hip_runtime.h>

typedef float v2f __attribute__((ext_vector_type(2)));
typedef float v8f __attribute__((ext_vector_type(8)));

#define NPTS   500000
#define NTERMS 64
#define NOUT   4
#define KSTEPS (NPTS / 4)

__device__ __forceinline__ constexpr int tri(int i, int j) {
  // upper-triangle linear index for 4x4 tile grid, i <= j
  return i * 4 + j - ((i * (i + 1)) >> 1);
}

// ---------------------------------------------------------------------------
// Kernel 0: zero the workspace accumulators (xtx: 64*64, xty: 64*4)
// ---------------------------------------------------------------------------
__global__ void zero_ws_kernel(float* __restrict__ ws) {
  int i = blockIdx.x * blockDim.x + threadIdx.x;
  if (i < NTERMS * NTERMS + NTERMS * NOUT) ws[i] = 0.0f;
}

// ---------------------------------------------------------------------------
// Kernel 1: fused  (copy X -> out)  +  (XtX partials)  +  (XtY partials)
// One pass over X. Each wave owns a grid-strided set of 4-row K-steps,
// accumulates 10 upper-tri Gram tiles + 4 XtY tiles in VGPRs via
// v_wmma_f32_16x16x4_f32, then reduces: VGPR -> LDS (ds_add_f32) ->
// global f32 atomics into d_ws.
// ---------------------------------------------------------------------------
__launch_bounds__(256)
__global__ void gram_xty_kernel(const float* __restrict__ X,
                                const float* __restrict__ Y,
                                float* __restrict__ Xcopy,
                                float* __restrict__ ws) {
  __shared__ float sG[NTERMS * NTERMS];   // 16 KB
  __shared__ float sY[NTERMS * NOUT];     // 1 KB
  const int tid = threadIdx.x;
  for (int i = tid; i < NTERMS * NTERMS; i += 256) sG[i] = 0.0f;
  for (int i = tid; i < NTERMS * NOUT; i += 256) sY[i] = 0.0f;
  __syncthreads();

  const int lane = tid & 31;
  const int wi   = lane >> 4;   // half-wave select: K rows {0,1} vs {2,3}
  const int c    = lane & 15;   // column within a 16-wide group
  const int gwave = blockIdx.x * (blockDim.x >> 5) + (tid >> 5);
  const int nwave = gridDim.x * (blockDim.x >> 5);

  v8f accG[10];
  v8f accY[4];
  const v8f vzero = {0.f, 0.f, 0.f, 0.f, 0.f, 0.f, 0.f, 0.f};
#pragma unroll
  for (int t = 0; t < 10; ++t) accG[t] = vzero;
#pragma unroll
  for (int t = 0; t < 4; ++t) accY[t] = vzero;

  for (int ks = gwave; ks < KSTEPS; ks += nwave) {
    const size_t r0 = (size_t)ks * 4 + 2 * wi;  // first of two K rows this half-wave holds
    const float* __restrict__ xp = X + r0 * NTERMS + c;
    float* __restrict__ op = Xcopy + r0 * NTERMS + c;

    // A-frag == B-frag for f32 16x16x4: lane L<16 holds K=0,1 ; lane 16+L holds K=2,3.
    v2f frag[4];
#pragma unroll
    for (int g = 0; g < 4; ++g) {
      float a = __builtin_nontemporal_load(xp + g * 16);
      float b = __builtin_nontemporal_load(xp + NTERMS + g * 16);
      v2f f;
      f.x = a;
      f.y = b;
      frag[g] = f;
      // fused identity copy of thetas into the output buffer
      __builtin_nontemporal_store(a, op + g * 16);
      __builtin_nontemporal_store(b, op + NTERMS + g * 16);
    }

    // B-fragment for Y (4 x 16, columns >= NOUT are zero)
    v2f yfrag;
    yfrag.x = 0.0f;
    yfrag.y = 0.0f;
    if (c < NOUT) {
      yfrag.x = Y[r0 * NOUT + c];
      yfrag.y = Y[(r0 + 1) * NOUT + c];
    }

#pragma unroll
    for (int i = 0; i < 4; ++i) {
#pragma unroll
      for (int j = i; j < 4; ++j) {
        accG[tri(i, j)] = __builtin_amdgcn_wmma_f32_16x16x4_f32(
            false, frag[i], false, frag[j], (short)0, accG[tri(i, j)], false, false);
      }
      accY[i] = __builtin_amdgcn_wmma_f32_16x16x4_f32(
          false, frag[i], false, yfrag, (short)0, accY[i], false, false);
    }
  }

  // C/D layout: VGPR v, lane L  ->  row = 8*(L>=16) + v, col = L&15 within the tile.
#pragma unroll
  for (int i = 0; i < 4; ++i) {
#pragma unroll
    for (int j = i; j < 4; ++j) {
      v8f t = accG[tri(i, j)];
#pragma unroll
      for (int v = 0; v < 8; ++v) {
        int m = 16 * i + 8 * wi + v;
        int n = 16 * j + c;
        atomicAdd(&sG[m * NTERMS + n], t[v]);
        if (i != j) atomicAdd(&sG[n * NTERMS + m], t[v]);  // symmetric mirror
      }
    }
  }
  if (c < NOUT) {
#pragma unroll
    for (int i = 0; i < 4; ++i) {
      v8f t = accY[i];
#pragma unroll
      for (int v = 0; v < 8; ++v) {
        int m = 16 * i + 8 * wi + v;
        atomicAdd(&sY[m * NOUT + c], t[v]);
      }
    }
  }
  __syncthreads();

  for (int i = tid; i < NTERMS * NTERMS; i += 256) unsafeAtomicAdd(&ws[i], sG[i]);
  for (int i = tid; i < NTERMS * NOUT; i += 256)
    unsafeAtomicAdd(&ws[NTERMS * NTERMS + i], sY[i]);
}

// ---------------------------------------------------------------------------
// Kernel 2: solve (XtX) C = (XtY) by Gauss-Jordan with partial pivoting on the
// augmented 64x68 system in LDS. One block, 64 threads (thread t owns row t).
// ---------------------------------------------------------------------------
__launch_bounds__(64)
__global__ void solve_kernel(const float* __restrict__ ws, float* __restrict__ coeffs) {
  __shared__ float A[NTERMS][72];  // 68 used, padded
  __shared__ int s_piv;
  __shared__ float s_ipv;
  const int t = threadIdx.x;  // 0..63

  for (int cc = 0; cc < NTERMS; ++cc) A[t][cc] = ws[t * NTERMS + cc];
  for (int o = 0; o < NOUT; ++o) A[t][NTERMS + o] = ws[NTERMS * NTERMS + t * NOUT + o];
  __syncthreads();

  for (int k = 0; k < NTERMS; ++k) {
    if (t == 0) {
      int p = k;
      float best = fabsf(A[k][k]);
      for (int r = k + 1; r < NTERMS; ++r) {
        float v = fabsf(A[r][k]);
        if (v > best) { best = v; p = r; }
      }
      s_piv = p;
    }
    __syncthreads();
    const int p = s_piv;
    if (p != k) {
      for (int cc = t; cc < NTERMS + NOUT; cc += 64) {
        float tmp = A[k][cc];
        A[k][cc] = A[p][cc];
        A[p][cc] = tmp;
      }
    }
    __syncthreads();
    if (t == 0) s_ipv = 1.0f / A[k][k];
    __syncthreads();
    const float ipv = s_ipv;
    for (int cc = t; cc < NTERMS + NOUT; cc += 64) A[k][cc] *= ipv;
    __syncthreads();
    if (t != k) {
      const float f = A[t][k];
      for (int cc = 0; cc < NTERMS + NOUT; ++cc) A[t][cc] -= f * A[k][cc];
    }
    __syncthreads();
  }

  for (int o = 0; o < NOUT; ++o) coeffs[t * NOUT + o] = A[t][NTERMS + o];
}

// ---------------------------------------------------------------------------
extern "C" void kernel_launch(void* const* d_in, const int* in_sizes, int n_in,
                              void* d_out, int out_size, void* d_ws, size_t ws_size,
                              hipStream_t stream) {
  const float* X = (const float*)d_in[0];  // thetas [500000, 64] f32
  const float* Y = (const float*)d_in[1];  // time_derivs [500000, 4] f32
  float* out = (float*)d_out;
  float* ws = (float*)d_ws;

  float* Xcopy = out;                                     // first output: sparse_theta
  float* coeffs = out + (size_t)NPTS * NTERMS;            // second output: [64, 4]

  const int accum_elems = NTERMS * NTERMS + NTERMS * NOUT;  // 4352
  zero_ws_kernel<<<(accum_elems + 255) / 256, 256, 0, stream>>>(ws);

  // 512 blocks x 8 wave32 = 4096 waves; each wave handles ~31 K-steps (4 rows each).
  gram_xty_kernel<<<512, 256, 0, stream>>>(X, Y, Xcopy, ws);

  solve_kernel<<<1, 64, 0, stream>>>(ws, coeffs);
}